// AverageFusion_63462436765886
// MI455X (gfx1250) — compile-verified
//
#include <hip/hip_runtime.h>

// AverageFusion for MI455X (gfx1250, wave32).
// B=16384 rows, L=1024 f32 features, 4 modalities.
// Memory-bound: 320 MB traffic -> ~13.7 us floor at 23.3 TB/s.
// This version exercises the CDNA5 async copy path: the 16 KB row tile
// (4 modalities x 4 KB) is staged into LDS with GLOBAL_LOAD_ASYNC_TO_LDS_B128
// (ASYNCcnt-tracked), then consumed from LDS. HBM-bound either way; LDS
// bandwidth (256 B/clk/WGP) dwarfs per-WGP HBM share (~36 B/clk).

typedef float v4f __attribute__((ext_vector_type(4)));

#define ROW_LEN   1024           // L
#define BLOCK     256            // 8 wave32s; each thread owns one float4 per modality
#define WAVES     (BLOCK / 32)

// Low 32 bits of a generic pointer into LDS == addrspace(3) byte offset,
// which is what the async-load VDST operand expects (HW adds LDS_BASE).
static __device__ __forceinline__ unsigned lds_byte_off(const void* p) {
    return (unsigned)(unsigned long long)p;
}

static __device__ __forceinline__ void async_load_b128_to_lds(unsigned lds_off,
                                                              const void* gaddr) {
    asm volatile("global_load_async_to_lds_b128 %0, %1, off th:TH_LOAD_NT"
                 :: "v"(lds_off), "v"((unsigned long long)gaddr)
                 : "memory");
}

static __device__ __forceinline__ float wave32_reduce_add(float v) {
    // wave32 butterfly reduction (warpSize == 32 on gfx1250)
    v += __shfl_xor(v, 16, 32);
    v += __shfl_xor(v,  8, 32);
    v += __shfl_xor(v,  4, 32);
    v += __shfl_xor(v,  2, 32);
    v += __shfl_xor(v,  1, 32);
    return v;
}

__global__ __launch_bounds__(BLOCK, 8)
void average_fusion_kernel(const float* __restrict__ x0,
                           const float* __restrict__ x1,
                           const float* __restrict__ x2,
                           const float* __restrict__ x3,
                           float* __restrict__ out)
{
    __shared__ v4f   stage[4][BLOCK];    // 16 KB staged tile: 4 modalities x 1 row
    __shared__ float s_part[WAVES][4];   // per-wave partial sums, 4 modalities
    __shared__ float s_scaler;

    const int tid  = threadIdx.x;
    const int lane = tid & 31;
    const int wave = tid >> 5;

    // float4-granular index: row*256 + tid  (256 threads * 4 floats = 1024 = L)
    const size_t base = (size_t)blockIdx.x * (ROW_LEN / 4) + (size_t)tid;

    // Kick off four independent async B128 copies into LDS (ASYNCcnt path).
    async_load_b128_to_lds(lds_byte_off(&stage[0][tid]), (const v4f*)x0 + base);
    async_load_b128_to_lds(lds_byte_off(&stage[1][tid]), (const v4f*)x1 + base);
    async_load_b128_to_lds(lds_byte_off(&stage[2][tid]), (const v4f*)x2 + base);
    async_load_b128_to_lds(lds_byte_off(&stage[3][tid]), (const v4f*)x3 + base);

    // Each lane consumes only the 16 B per modality that it staged itself,
    // so the per-wave ASYNCcnt wait is sufficient ordering (no barrier needed).
    asm volatile("s_wait_asynccnt 0x0" ::: "memory");

    const v4f a = stage[0][tid];
    const v4f b = stage[1][tid];
    const v4f c = stage[2][tid];
    const v4f d = stage[3][tid];

    // Fused elementwise sum stays in registers until the scaler is known.
    v4f fused = a + b + c + d;

    // Per-thread partial row-sums for each modality.
    float sa = (a.x + a.y) + (a.z + a.w);
    float sb = (b.x + b.y) + (b.z + b.w);
    float sc = (c.x + c.y) + (c.z + c.w);
    float sd = (d.x + d.y) + (d.z + d.w);

    sa = wave32_reduce_add(sa);
    sb = wave32_reduce_add(sb);
    sc = wave32_reduce_add(sc);
    sd = wave32_reduce_add(sd);

    if (lane == 0) {
        s_part[wave][0] = sa;
        s_part[wave][1] = sb;
        s_part[wave][2] = sc;
        s_part[wave][3] = sd;
    }
    __syncthreads();

    if (tid == 0) {
        float t0 = 0.f, t1 = 0.f, t2 = 0.f, t3 = 0.f;
        #pragma unroll
        for (int w = 0; w < WAVES; ++w) {
            t0 += s_part[w][0];
            t1 += s_part[w][1];
            t2 += s_part[w][2];
            t3 += s_part[w][3];
        }
        const int zc = (t0 == 0.0f) + (t1 == 0.0f) + (t2 == 0.0f) + (t3 == 0.0f);
        s_scaler = (zc > 0) ? (float)(zc + 1) : 1.0f;
    }
    __syncthreads();

    const float scaler = s_scaler;
    fused *= scaler;

    __builtin_nontemporal_store(fused, (v4f*)out + base);
}

extern "C" void kernel_launch(void* const* d_in, const int* in_sizes, int n_in,
                              void* d_out, int out_size, void* d_ws, size_t ws_size,
                              hipStream_t stream) {
    (void)n_in; (void)out_size; (void)d_ws; (void)ws_size;
    const float* x0 = (const float*)d_in[0];
    const float* x1 = (const float*)d_in[1];
    const float* x2 = (const float*)d_in[2];
    const float* x3 = (const float*)d_in[3];
    float* out = (float*)d_out;

    const int rows = in_sizes[0] / ROW_LEN;   // B = 16384
    average_fusion_kernel<<<rows, BLOCK, 0, stream>>>(x0, x1, x2, x3, out);
}